// Graph_EGNN_56169582297514
// MI455X (gfx1250) — compile-verified
//
#include <hip/hip_runtime.h>
#include <hip/hip_bf16.h>
#include <math.h>

typedef float v2f __attribute__((ext_vector_type(2)));
typedef float v8f __attribute__((ext_vector_type(8)));

#define WAVES_PER_BLOCK 8

__device__ __forceinline__ v8f wmma4(v2f a, v2f b, v8f c) {
  // D(16x16,f32) = A(16x4,f32) x B(4x16,f32) + C ; full fp32 precision
  return __builtin_amdgcn_wmma_f32_16x16x4_f32(false, a, false, b, (short)0, c, false, false);
}

__device__ __forceinline__ float silu_f(float x) {
  return x / (1.0f + __expf(-x));
}

__device__ __forceinline__ void atomic_add_f32(float* p, float v) {
  __hip_atomic_fetch_add(p, v, __ATOMIC_RELAXED, __HIP_MEMORY_SCOPE_AGENT);
}

// ---------------------------------------------------------------------------
// workspace zeroing (harness poisons ws once; we must clear accumulators
// at the start of every launch so replays are deterministic)
// ---------------------------------------------------------------------------
__global__ void __launch_bounds__(256) egnn_zero_kernel(float* __restrict__ p, long n) {
  long i = (long)blockIdx.x * blockDim.x + threadIdx.x;
  if (i < n) p[i] = 0.0f;
}

// ---------------------------------------------------------------------------
// Edge kernel: per 16-edge tile (one tile per wave, grid-stride persistent)
//   f[76] = [h_src(32), h_dst(32), radial, edge_feat(8), pad(3)]
//   h1 = silu(f @ We1 + be1); msg = silu(h1 @ We2 + be2)
//   gate = silu(msg @ Wc1 + bc1) @ Wc2
//   atomics: h_accum[dst] += msg ; x_accum[dst] += gate * unit(x_diff); deg[dst]++
// ---------------------------------------------------------------------------
struct alignas(16) EdgeSmem {
  float Wt1[32][76];   // We1 column-major, K padded 73->76 with zeros
  float Wt2[32][32];   // We2 column-major
  float Wtc1[32][32];  // Wc1 column-major
  float be1s[32], be2s[32], bc1s[32], wc2s[32];
  float stage[WAVES_PER_BLOCK][16 * 32];  // per-wave 16x32 transpose staging
  float gate[WAVES_PER_BLOCK][16];
};

__global__ void __launch_bounds__(256) egnn_edge_kernel(
    const float* __restrict__ node_feat, const float* __restrict__ coord,
    const float* __restrict__ edge_feat, const int* __restrict__ src,
    const int* __restrict__ dst,
    const float* __restrict__ We1, const float* __restrict__ be1,
    const float* __restrict__ We2, const float* __restrict__ be2,
    const float* __restrict__ Wc1, const float* __restrict__ bc1,
    const float* __restrict__ Wc2,
    float* __restrict__ h_accum, float* __restrict__ x_accum,
    float* __restrict__ deg, long E)
{
  __shared__ EdgeSmem s;
  const int tid = threadIdx.x;
  for (int i = tid; i < 32 * 76; i += 256) {
    int col = i / 76, k = i - col * 76;
    s.Wt1[col][k] = (k < 73) ? We1[k * 32 + col] : 0.0f;
  }
  for (int i = tid; i < 32 * 32; i += 256) {
    int col = i >> 5, k = i & 31;
    s.Wt2[col][k]  = We2[k * 32 + col];
    s.Wtc1[col][k] = Wc1[k * 32 + col];
  }
  if (tid < 32) {
    s.be1s[tid] = be1[tid]; s.be2s[tid] = be2[tid];
    s.bc1s[tid] = bc1[tid]; s.wc2s[tid] = Wc2[tid];
  }
  __syncthreads();

  const int lane = tid & 31;
  const int wave = tid >> 5;
  const int M   = lane & 15;   // row (A/D) or column (B/D) index
  const int h2  = lane >> 4;   // lane half
  const int k0h = 2 * h2;      // K offset within a K=4 tile
  float* const st = s.stage[wave];

  const long numTiles = (E + 15) >> 4;
  const long stride   = (long)gridDim.x * WAVES_PER_BLOCK;

  for (long tile = (long)blockIdx.x * WAVES_PER_BLOCK + wave; tile < numTiles; tile += stride) {
    const long e     = tile * 16 + M;
    const bool valid = (e < E);
    const long ec    = valid ? e : (E - 1);
    const int  se    = src[ec];
    const int  de    = dst[ec];

    // prefetch next tile's streaming edge data into cache
    {
      long nt = tile + stride;
      if (nt < numTiles) {
        __builtin_prefetch(edge_feat + (nt * 16 + M) * 8, 0, 0);
        __builtin_prefetch(src + nt * 16 + M, 0, 0);
        __builtin_prefetch(dst + nt * 16 + M, 0, 0);
      }
    }

    // radial + unit direction (redundant across lane halves, cheap)
    const float csx = coord[(long)se * 3 + 0];
    const float csy = coord[(long)se * 3 + 1];
    const float csz = coord[(long)se * 3 + 2];
    const float cdx = coord[(long)de * 3 + 0];
    const float cdy = coord[(long)de * 3 + 1];
    const float cdz = coord[(long)de * 3 + 2];
    const float dxv = csx - cdx, dyv = csy - cdy, dzv = csz - cdz;
    const float radial = dxv * dxv + dyv * dyv + dzv * dzv;
    const float inv = 1.0f / (sqrtf(radial) + 1e-30f);
    const float ux = dxv * inv, uy = dyv * inv, uz = dzv * inv;

    const float4 efa = *(const float4*)(edge_feat + ec * 8);
    const float4 efb = *(const float4*)(edge_feat + ec * 8 + 4);

    // ---------------- edge MLP layer 1: [16 x 76] @ [76 x 32] ----------------
    v8f a0 = {0,0,0,0,0,0,0,0};
    v8f a1 = {0,0,0,0,0,0,0,0};
    const float* srow = node_feat + (long)se * 32;
    const float* drow = node_feat + (long)de * 32;
#pragma unroll
    for (int t = 0; t < 8; ++t) {  // K = 0..31 : h_src
      v2f A = *(const v2f*)(srow + 4 * t + k0h);
      a0 = wmma4(A, *(const v2f*)&s.Wt1[M][4 * t + k0h], a0);
      a1 = wmma4(A, *(const v2f*)&s.Wt1[16 + M][4 * t + k0h], a1);
    }
#pragma unroll
    for (int t = 0; t < 8; ++t) {  // K = 32..63 : h_dst
      v2f A = *(const v2f*)(drow + 4 * t + k0h);
      a0 = wmma4(A, *(const v2f*)&s.Wt1[M][32 + 4 * t + k0h], a0);
      a1 = wmma4(A, *(const v2f*)&s.Wt1[16 + M][32 + 4 * t + k0h], a1);
    }
    {  // K = 64..75 : radial, edge_feat[0..7], zero pad
      v2f A;
      A[0] = h2 ? efa.y : radial; A[1] = h2 ? efa.z : efa.x;
      a0 = wmma4(A, *(const v2f*)&s.Wt1[M][64 + k0h], a0);
      a1 = wmma4(A, *(const v2f*)&s.Wt1[16 + M][64 + k0h], a1);
      A[0] = h2 ? efb.y : efa.w;  A[1] = h2 ? efb.z : efb.x;
      a0 = wmma4(A, *(const v2f*)&s.Wt1[M][68 + k0h], a0);
      a1 = wmma4(A, *(const v2f*)&s.Wt1[16 + M][68 + k0h], a1);
      A[0] = h2 ? 0.0f : efb.w;   A[1] = 0.0f;
      a0 = wmma4(A, *(const v2f*)&s.Wt1[M][72 + k0h], a0);
      a1 = wmma4(A, *(const v2f*)&s.Wt1[16 + M][72 + k0h], a1);
    }

    // bias + SiLU, stage to LDS in row-major for next A fragments
    {
      const float b0 = s.be1s[M], b1 = s.be1s[16 + M];
#pragma unroll
      for (int r = 0; r < 8; ++r) {
        const int row = r + 8 * h2;
        st[row * 32 + M]      = silu_f(a0[r] + b0);
        st[row * 32 + 16 + M] = silu_f(a1[r] + b1);
      }
    }
    __builtin_amdgcn_wave_barrier();  // same-wave LDS is in-order; block compiler reordering

    // ---------------- edge MLP layer 2: [16 x 32] @ [32 x 32] ----------------
    a0 = (v8f){0,0,0,0,0,0,0,0};
    a1 = (v8f){0,0,0,0,0,0,0,0};
#pragma unroll
    for (int t = 0; t < 8; ++t) {
      v2f A = *(const v2f*)&st[M * 32 + 4 * t + k0h];
      a0 = wmma4(A, *(const v2f*)&s.Wt2[M][4 * t + k0h], a0);
      a1 = wmma4(A, *(const v2f*)&s.Wt2[16 + M][4 * t + k0h], a1);
    }
    float m0[8], m1[8];
    {
      const float b0 = s.be2s[M], b1 = s.be2s[16 + M];
#pragma unroll
      for (int r = 0; r < 8; ++r) {
        m0[r] = silu_f(a0[r] + b0);
        m1[r] = silu_f(a1[r] + b1);
      }
    }

    // scatter h messages (f32 atomics into L2-resident accumulator)
#pragma unroll
    for (int r = 0; r < 8; ++r) {
      const int  row = r + 8 * h2;
      const int  dd  = __shfl(de, row, 32);
      if (tile * 16 + row < E) {
        atomic_add_f32(&h_accum[(long)dd * 32 + M],      m0[r]);
        atomic_add_f32(&h_accum[(long)dd * 32 + 16 + M], m1[r]);
      }
    }

    // stage msg for the coord MLP A fragments
    __builtin_amdgcn_wave_barrier();
#pragma unroll
    for (int r = 0; r < 8; ++r) {
      const int row = r + 8 * h2;
      st[row * 32 + M]      = m0[r];
      st[row * 32 + 16 + M] = m1[r];
    }
    __builtin_amdgcn_wave_barrier();

    // ---------------- coord MLP layer 1: [16 x 32] @ [32 x 32] ---------------
    a0 = (v8f){0,0,0,0,0,0,0,0};
    a1 = (v8f){0,0,0,0,0,0,0,0};
#pragma unroll
    for (int t = 0; t < 8; ++t) {
      v2f A = *(const v2f*)&st[M * 32 + 4 * t + k0h];
      a0 = wmma4(A, *(const v2f*)&s.Wtc1[M][4 * t + k0h], a0);
      a1 = wmma4(A, *(const v2f*)&s.Wtc1[16 + M][4 * t + k0h], a1);
    }

    // gate[row] = sum_n silu(hc[row][n] + bc1[n]) * Wc2[n] : lane-butterfly reduce
    {
      const float b0 = s.bc1s[M], b1 = s.bc1s[16 + M];
      const float w0 = s.wc2s[M], w1 = s.wc2s[16 + M];
      float p[8];
#pragma unroll
      for (int r = 0; r < 8; ++r)
        p[r] = silu_f(a0[r] + b0) * w0 + silu_f(a1[r] + b1) * w1;
#pragma unroll
      for (int r = 0; r < 8; ++r) {
        p[r] += __shfl_xor(p[r], 1, 32);
        p[r] += __shfl_xor(p[r], 2, 32);
        p[r] += __shfl_xor(p[r], 4, 32);
        p[r] += __shfl_xor(p[r], 8, 32);
      }
#pragma unroll
      for (int r = 0; r < 8; ++r)
        if (M == r) s.gate[wave][r + 8 * h2] = p[r];
    }
    __builtin_amdgcn_wave_barrier();

    // scatter coord messages + degree (one lane half per edge)
    if (h2 == 0 && valid) {
      const float g  = s.gate[wave][M];
      const long  b3 = (long)de * 3;
      atomic_add_f32(&x_accum[b3 + 0], g * ux);
      atomic_add_f32(&x_accum[b3 + 1], g * uy);
      atomic_add_f32(&x_accum[b3 + 2], g * uz);
      atomic_add_f32(&deg[de], 1.0f);
    }
    __builtin_amdgcn_wave_barrier();
  }
}

// ---------------------------------------------------------------------------
// Node kernel: h = silu([h_in, h_neigh] @ Wn1 + bn1) @ Wn2 + bn2
//              x = coord + x_accum / max(deg, 1)
// ---------------------------------------------------------------------------
struct alignas(16) NodeSmem {
  float Wn1t[32][64];  // Wn1 column-major
  float Wn2t[32][32];  // Wn2 column-major
  float bn1s[32], bn2s[32];
  float stage[WAVES_PER_BLOCK][16 * 32];
};

__global__ void __launch_bounds__(256) egnn_node_kernel(
    const float* __restrict__ node_feat, const float* __restrict__ coord,
    const float* __restrict__ Wn1, const float* __restrict__ bn1,
    const float* __restrict__ Wn2, const float* __restrict__ bn2,
    const float* __restrict__ h_accum, const float* __restrict__ x_accum,
    const float* __restrict__ deg, float* __restrict__ out, long N)
{
  __shared__ NodeSmem s;
  const int tid = threadIdx.x;
  for (int i = tid; i < 32 * 64; i += 256) { int col = i >> 6, k = i & 63; s.Wn1t[col][k] = Wn1[k * 32 + col]; }
  for (int i = tid; i < 32 * 32; i += 256) { int col = i >> 5, k = i & 31; s.Wn2t[col][k] = Wn2[k * 32 + col]; }
  if (tid < 32) { s.bn1s[tid] = bn1[tid]; s.bn2s[tid] = bn2[tid]; }
  __syncthreads();

  const int lane = tid & 31, wave = tid >> 5;
  const int M = lane & 15, h2 = lane >> 4, k0h = 2 * h2;
  float* const st = s.stage[wave];

  const long numTiles = (N + 15) >> 4;
  const long stride   = (long)gridDim.x * WAVES_PER_BLOCK;
  for (long tile = (long)blockIdx.x * WAVES_PER_BLOCK + wave; tile < numTiles; tile += stride) {
    const long node  = tile * 16 + M;
    const bool valid = (node < N);
    const long nc    = valid ? node : (N - 1);

    v8f a0 = {0,0,0,0,0,0,0,0};
    v8f a1 = {0,0,0,0,0,0,0,0};
    const float* fr = node_feat + nc * 32;
    const float* hr = h_accum + nc * 32;
#pragma unroll
    for (int t = 0; t < 8; ++t) {  // K = 0..31 : node_feat
      v2f A = *(const v2f*)(fr + 4 * t + k0h);
      a0 = wmma4(A, *(const v2f*)&s.Wn1t[M][4 * t + k0h], a0);
      a1 = wmma4(A, *(const v2f*)&s.Wn1t[16 + M][4 * t + k0h], a1);
    }
#pragma unroll
    for (int t = 0; t < 8; ++t) {  // K = 32..63 : h_neigh
      v2f A = *(const v2f*)(hr + 4 * t + k0h);
      a0 = wmma4(A, *(const v2f*)&s.Wn1t[M][32 + 4 * t + k0h], a0);
      a1 = wmma4(A, *(const v2f*)&s.Wn1t[16 + M][32 + 4 * t + k0h], a1);
    }
    {
      const float b0 = s.bn1s[M], b1 = s.bn1s[16 + M];
#pragma unroll
      for (int r = 0; r < 8; ++r) {
        const int row = r + 8 * h2;
        st[row * 32 + M]      = silu_f(a0[r] + b0);
        st[row * 32 + 16 + M] = silu_f(a1[r] + b1);
      }
    }
    __builtin_amdgcn_wave_barrier();

    a0 = (v8f){0,0,0,0,0,0,0,0};
    a1 = (v8f){0,0,0,0,0,0,0,0};
#pragma unroll
    for (int t = 0; t < 8; ++t) {
      v2f A = *(const v2f*)&st[M * 32 + 4 * t + k0h];
      a0 = wmma4(A, *(const v2f*)&s.Wn2t[M][4 * t + k0h], a0);
      a1 = wmma4(A, *(const v2f*)&s.Wn2t[16 + M][4 * t + k0h], a1);
    }
    {
      const float b0 = s.bn2s[M], b1 = s.bn2s[16 + M];
#pragma unroll
      for (int r = 0; r < 8; ++r) {
        const long row = tile * 16 + r + 8 * h2;
        if (row < N) {
          out[row * 32 + M]      = a0[r] + b0;
          out[row * 32 + 16 + M] = a1[r] + b1;
        }
      }
    }
    if (h2 == 0 && valid) {
      const float invd = 1.0f / fmaxf(deg[node], 1.0f);
      const long  b3   = node * 3;
      out[N * 32 + b3 + 0] = coord[b3 + 0] + x_accum[b3 + 0] * invd;
      out[N * 32 + b3 + 1] = coord[b3 + 1] + x_accum[b3 + 1] * invd;
      out[N * 32 + b3 + 2] = coord[b3 + 2] + x_accum[b3 + 2] * invd;
    }
    __builtin_amdgcn_wave_barrier();
  }
}

// ---------------------------------------------------------------------------
extern "C" void kernel_launch(void* const* d_in, const int* in_sizes, int n_in,
                              void* d_out, int out_size, void* d_ws, size_t ws_size,
                              hipStream_t stream) {
  (void)n_in; (void)out_size; (void)ws_size;
  const float* node_feat = (const float*)d_in[0];
  const float* coord     = (const float*)d_in[1];
  const float* edge_feat = (const float*)d_in[2];
  const int*   src       = (const int*)d_in[3];
  const int*   dst       = (const int*)d_in[4];
  const float* We1 = (const float*)d_in[5];
  const float* be1 = (const float*)d_in[6];
  const float* We2 = (const float*)d_in[7];
  const float* be2 = (const float*)d_in[8];
  const float* Wn1 = (const float*)d_in[9];
  const float* bn1 = (const float*)d_in[10];
  const float* Wn2 = (const float*)d_in[11];
  const float* bn2 = (const float*)d_in[12];
  const float* Wc1 = (const float*)d_in[13];
  const float* bc1 = (const float*)d_in[14];
  const float* Wc2 = (const float*)d_in[15];

  const long Nn = in_sizes[0] / 32;
  const long Ee = in_sizes[3];

  float* h_accum = (float*)d_ws;                 // [N, 32]
  float* x_accum = h_accum + (size_t)Nn * 32;    // [N, 3]
  float* degp    = x_accum + (size_t)Nn * 3;     // [N]
  float* out     = (float*)d_out;                // h [N,32] then x [N,3]

  const long zc = Nn * 36;
  egnn_zero_kernel<<<(int)((zc + 255) / 256), 256, 0, stream>>>(h_accum, zc);

  const long eTiles  = (Ee + 15) / 16;
  long       eBlocks = (eTiles + WAVES_PER_BLOCK - 1) / WAVES_PER_BLOCK;
  if (eBlocks > 4096) eBlocks = 4096;
  egnn_edge_kernel<<<(int)eBlocks, 256, 0, stream>>>(
      node_feat, coord, edge_feat, src, dst,
      We1, be1, We2, be2, Wc1, bc1, Wc2,
      h_accum, x_accum, degp, Ee);

  const long nTiles  = (Nn + 15) / 16;
  long       nBlocks = (nTiles + WAVES_PER_BLOCK - 1) / WAVES_PER_BLOCK;
  if (nBlocks > 2048) nBlocks = 2048;
  egnn_node_kernel<<<(int)nBlocks, 256, 0, stream>>>(
      node_feat, coord, Wn1, bn1, Wn2, bn2,
      h_accum, x_accum, degp, out, Nn);
}